// PlasticNet_47742856462700
// MI455X (gfx1250) — compile-verified
//
#include <hip/hip_runtime.h>
#include <hip/hip_bf16.h>
#include <math.h>

// Problem constants (from reference)
#define TT 64
#define BB 32
#define II 128
#define HH 256
#define CLIPV 2.0f

typedef __attribute__((ext_vector_type(2))) float v2f;
typedef __attribute__((ext_vector_type(8))) float v8f;

// ---------------------------------------------------------------------------
// Kernel 1: precompute pre-activations  Y[t*B+b, k] = x[t,b,:] @ Wi[:,k] + bi[k]
// GEMM: [M=2048, K=128] x [K=128, N=256] in f32 via V_WMMA_F32_16X16X4_F32.
// One 16x16 output tile per wave; 8 waves (256 thr) per block; 256 blocks.
//
// VGPR layouts per CDNA5 ISA 7.12.2 (wave32):
//  A 16x4 f32 : lane l -> M = l%16, K = (l/16)*2 + {vgpr0,vgpr1}
//  B  4x16 f32: lane l -> N = l%16, K = (l/16)*2 + {vgpr0,vgpr1}
//  C/D 16x16  : lane l, vgpr r -> M = r + 8*(l/16), N = l%16
// ---------------------------------------------------------------------------
__global__ __launch_bounds__(256) void xwi_gemm_wmma(
    const float* __restrict__ X,   // [2048,128]
    const float* __restrict__ Wi,  // [128,256]
    const float* __restrict__ bi,  // [256]
    float* __restrict__ Y)         // [2048,256]
{
  const int lane = threadIdx.x & 31;
  const int wave = threadIdx.x >> 5;
  const int tile = blockIdx.x * 8 + wave;      // 0..2047 tiles (128 Mtiles x 16 Ntiles)
  const int tileN = tile & 15;
  const int tileM = tile >> 4;

  const int mrow = tileM * 16 + (lane & 15);
  const int ncol = tileN * 16 + (lane & 15);
  const int koff = (lane >> 4) << 1;           // 0 or 2

  v8f c = {};
  #pragma unroll 4
  for (int k0 = 0; k0 < II; k0 += 4) {
    // A: two consecutive K values for this lane's row (8B aligned: k even)
    v2f a = *(const v2f*)(X + (size_t)mrow * II + k0 + koff);
    // B: two K rows of column ncol
    v2f bm;
    bm.x = Wi[(size_t)(k0 + koff) * HH + ncol];
    bm.y = Wi[(size_t)(k0 + koff + 1) * HH + ncol];
    c = __builtin_amdgcn_wmma_f32_16x16x4_f32(
            /*neg_a=*/false, a, /*neg_b=*/false, bm,
            /*c_mod=*/(short)0, c, /*reuse_a=*/false, /*reuse_b=*/false);
  }

  const float bn = bi[ncol];
  const int mbase = tileM * 16 + ((lane >> 4) << 3);
  #pragma unroll
  for (int r = 0; r < 8; ++r) {
    Y[(size_t)(mbase + r) * HH + ncol] = c[r] + bn;
  }
}

// ---------------------------------------------------------------------------
// Kernel 2: the recurrence. One workgroup per batch sample; hebb[H][H] lives
// in LDS (256KB of the 320KB WGP budget) for the entire T=64 loop.
// 1024 threads = 32 wave32s.
// ---------------------------------------------------------------------------
__global__ __launch_bounds__(1024) void plastic_rnn(
    const float* __restrict__ x,     // [T,B,I]  (fallback path only)
    const float* __restrict__ Wi,    // [I,H]    (fallback path only)
    const float* __restrict__ bi,    // [H]      (fallback path only)
    const float* __restrict__ w,     // [H,H]
    const float* __restrict__ alpha, // [H,H]
    const float* __restrict__ Wm,    // [H,1]
    const float* __restrict__ bm,    // [1]
    const float* __restrict__ Wf,    // [1,H]
    const float* __restrict__ bf,    // [H]
    const float* __restrict__ xWi,   // [T,B,H] precomputed (or nullptr)
    float* __restrict__ out)         // ys[T,B,H] | h_f[B,H] | hebb_f[B,H,H]
{
  const int b   = blockIdx.x;
  const int tid = threadIdx.x;

  extern __shared__ float smem[];
  float* hebb_s = smem;                 // H*H   = 65536 floats
  float* h_s    = smem + HH * HH;       // H
  float* hn_s   = h_s + HH;             // H
  float* me_s   = hn_s + HH;            // H
  float* red    = me_s + HH;            // 1024
  float* x_s    = red + 1024;           // I (fallback staging)

  // zero-init state
  for (int j = tid; j < HH * HH; j += 1024) hebb_s[j] = 0.0f;
  if (tid < HH) h_s[tid] = 0.0f;
  __syncthreads();

  const int k    = tid & (HH - 1);
  const int part = tid >> 8;            // 0..3 : quarter of the i-range

  float* ys    = out;
  float* hf    = out + (size_t)TT * BB * HH;
  float* hebbf = hf + (size_t)BB * HH;

  for (int t = 0; t < TT; ++t) {
    if (xWi == nullptr && tid < II)
      x_s[tid] = x[((size_t)t * BB + b) * II + tid];

    // ---- pass 1: rec[k] = sum_i h[i]*(w[i,k] + alpha[i,k]*hebb[i,k]) ----
    float acc = 0.0f;
    const int i0 = part * (HH / 4);
    #pragma unroll 4
    for (int i = i0; i < i0 + HH / 4; ++i) {
      const int idx = i * HH + k;
      acc = fmaf(h_s[i], fmaf(alpha[idx], hebb_s[idx], w[idx]), acc);
    }
    red[tid] = acc;
    __syncthreads();

    if (tid < HH) {
      float rec = red[k] + red[HH + k] + red[2 * HH + k] + red[3 * HH + k];
      float pre;
      if (xWi) {
        pre = xWi[((size_t)t * BB + b) * HH + k] + rec;
      } else {
        pre = bi[k] + rec;
        #pragma unroll 4
        for (int i = 0; i < II; ++i) pre = fmaf(x_s[i], Wi[(size_t)i * HH + k], pre);
      }
      float hn = tanhf(pre);
      hn_s[k] = hn;
      ys[((size_t)t * BB + b) * HH + k] = hn;
    }
    __syncthreads();

    // ---- eta = tanh(h' . Wm + bm); myeta[k] = eta*Wf[k] + bf[k] ----
    if (tid < HH) red[tid] = hn_s[tid] * Wm[tid];
    __syncthreads();
    #pragma unroll
    for (int s = HH / 2; s > 0; s >>= 1) {
      if (tid < s) red[tid] += red[tid + s];
      __syncthreads();
    }
    if (tid < HH) {
      float eta = tanhf(red[0] + bm[0]);
      me_s[k] = fmaf(eta, Wf[k], bf[k]);
    }
    __syncthreads();

    // ---- pass 2: hebb[i,k] = clip(hebb + myeta[k]*h_old[i]*h'[k], +/-2) ----
    #pragma unroll 4
    for (int j = 0; j < (HH * HH) / 1024; ++j) {
      const int idx = tid + j * 1024;
      const int i   = idx >> 8;
      const int kk  = idx & (HH - 1);
      float v = fmaf(me_s[kk] * h_s[i], hn_s[kk], hebb_s[idx]);
      v = fminf(CLIPV, fmaxf(-CLIPV, v));
      hebb_s[idx] = v;
    }
    __syncthreads();

    if (tid < HH) h_s[tid] = hn_s[tid];
    __syncthreads();
  }

  // final state writeback
  if (tid < HH) hf[(size_t)b * HH + tid] = h_s[tid];
  for (int j = tid; j < HH * HH; j += 1024)
    hebbf[(size_t)b * HH * HH + j] = hebb_s[j];
}

extern "C" void kernel_launch(void* const* d_in, const int* in_sizes, int n_in,
                              void* d_out, int out_size, void* d_ws, size_t ws_size,
                              hipStream_t stream) {
  const float* x     = (const float*)d_in[0];
  const float* Wi    = (const float*)d_in[1];
  const float* bi    = (const float*)d_in[2];
  const float* w     = (const float*)d_in[3];
  const float* alpha = (const float*)d_in[4];
  const float* Wm    = (const float*)d_in[5];
  const float* bm    = (const float*)d_in[6];
  const float* Wf    = (const float*)d_in[7];
  const float* bf    = (const float*)d_in[8];
  float* out = (float*)d_out;

  const size_t need = (size_t)TT * BB * HH * sizeof(float);  // 2 MB
  const float* xWi = nullptr;
  if (ws_size >= need) {
    xWi = (const float*)d_ws;
    // 2048 M-rows / 16 = 128 M-tiles; 256/16 = 16 N-tiles; 2048 tiles / 8 waves
    xwi_gemm_wmma<<<256, 256, 0, stream>>>(x, Wi, bi, (float*)d_ws);
  }

  const size_t smem = (size_t)(HH * HH + 3 * HH + 1024 + II) * sizeof(float); // 269,824 B
  plastic_rnn<<<BB, 1024, smem, stream>>>(x, Wi, bi, w, alpha, Wm, bm, Wf, bf, xWi, out);
}